// Hallucigraph_40973988004063
// MI455X (gfx1250) — compile-verified
//
#include <hip/hip_runtime.h>
#include <math.h>

typedef __attribute__((ext_vector_type(2))) float v2f;
typedef __attribute__((ext_vector_type(8))) float v8f;

#define N_NODES 100000
#define N_EDGES 1600000
#define NFEAT 500
#define NHID 128
#define NCLASS 40
#define NCPAD 48

// ---------------------------------------------------------------------------
// GEMM1: support1[100000x128] = x[100000x500] @ W1[500x128]
// One block = one 16-row tile; 8 waves each own one 16-col tile of NHID.
// A tile staged in LDS once per block (shared by all 8 waves).
// x is a 200 MB single-use stream -> NT loads so it does not evict the
// L2-resident support/agg working set.
// ---------------------------------------------------------------------------
__global__ __launch_bounds__(256) void gcn_gemm1(const float* __restrict__ x,
                                                 const float* __restrict__ W1,
                                                 float* __restrict__ out) {
  __shared__ float ldsA[16 * NFEAT];  // 32 KB
  const int tid = threadIdx.x;
  const int lane = tid & 31;
  const int wave = tid >> 5;  // 0..7 -> column tile
  const int row0 = blockIdx.x * 16;

  for (int i = tid; i < 16 * NFEAT; i += 256) {
    int r = i / NFEAT;
    int c = i - r * NFEAT;
    ldsA[i] = __builtin_nontemporal_load(&x[(size_t)(row0 + r) * NFEAT + c]);
  }
  __syncthreads();

  const int m = lane & 15;
  const int kh = (lane >> 4) << 1;  // 0 for lanes 0-15, 2 for lanes 16-31
  const int col0 = wave * 16;
  const float* Arow = ldsA + m * NFEAT + kh;
  const float* Bp = W1 + col0 + m;  // N index == lane&15

  v8f acc = {};
  for (int k = 0; k < NFEAT; k += 4) {
    v2f a, b;
    a.x = Arow[k];
    a.y = Arow[k + 1];
    b.x = Bp[(size_t)(k + kh) * NHID];
    b.y = Bp[(size_t)(k + kh + 1) * NHID];
    acc = __builtin_amdgcn_wmma_f32_16x16x4_f32(false, a, false, b, (short)0,
                                                acc, false, false);
  }

  const int mb = (lane < 16) ? 0 : 8;
  float* op = out + (size_t)(row0 + mb) * NHID + col0 + m;
#pragma unroll
  for (int v = 0; v < 8; ++v) op[(size_t)v * NHID] = acc[v];
}

// ---------------------------------------------------------------------------
// GEMM2: support2[100000x40] = relu(agg1)[100000x128] @ W2[128x40]
// One block = one 16-row tile; 3 waves each own one 16-col tile of NCPAD=48.
// ReLU fused into the LDS stage of A; B zero-padded to 48 cols.
// agg1 is last-use here -> NT loads free L2 for the layer-2 working set.
// ---------------------------------------------------------------------------
__global__ __launch_bounds__(96) void gcn_gemm2(const float* __restrict__ agg1,
                                                const float* __restrict__ W2,
                                                float* __restrict__ out) {
  __shared__ float ldsA[16 * NHID];     // 8 KB
  __shared__ float ldsB[NHID * NCPAD];  // 24 KB
  const int tid = threadIdx.x;
  const int lane = tid & 31;
  const int wave = tid >> 5;  // 0..2 -> column tile
  const int row0 = blockIdx.x * 16;

  for (int i = tid; i < 16 * NHID; i += 96) {
    int r = i >> 7;
    int c = i & (NHID - 1);
    float v =
        __builtin_nontemporal_load(&agg1[(size_t)(row0 + r) * NHID + c]);
    ldsA[i] = v > 0.0f ? v : 0.0f;  // fused ReLU
  }
  for (int i = tid; i < NHID * NCPAD; i += 96) {
    int r = i / NCPAD;
    int c = i - r * NCPAD;
    ldsB[i] = (c < NCLASS) ? W2[r * NCLASS + c] : 0.0f;
  }
  __syncthreads();

  const int m = lane & 15;
  const int kh = (lane >> 4) << 1;
  const int col0 = wave * 16;

  v8f acc = {};
  for (int k = 0; k < NHID; k += 4) {
    v2f a, b;
    a.x = ldsA[m * NHID + k + kh];
    a.y = ldsA[m * NHID + k + kh + 1];
    b.x = ldsB[(k + kh) * NCPAD + col0 + m];
    b.y = ldsB[(k + kh + 1) * NCPAD + col0 + m];
    acc = __builtin_amdgcn_wmma_f32_16x16x4_f32(false, a, false, b, (short)0,
                                                acc, false, false);
  }

  const int mb = (lane < 16) ? 0 : 8;
  const int n = col0 + m;
  if (n < NCLASS) {
#pragma unroll
    for (int v = 0; v < 8; ++v)
      out[(size_t)(row0 + mb + v) * NCLASS + n] = acc[v];
  }
}

// ---------------------------------------------------------------------------
// agg[i] = bias[i % ncols]  (broadcast bias; re-initialized every call)
// ---------------------------------------------------------------------------
__global__ void gcn_bias_init(const float* __restrict__ b,
                              float* __restrict__ agg, int ncols, long total) {
  long i = (long)blockIdx.x * blockDim.x + threadIdx.x;
  if (i < total) agg[i] = b[i % ncols];
}

// ---------------------------------------------------------------------------
// Scatter layer 1: one wave per edge; lane handles 4 of 128 features.
// Gather is a float4 (global_load_b128) kept RT (L2-hot); adds are
// global_atomic_add_f32 on the L2-resident agg. Edge arrays are streamed
// once -> NT loads.
// ---------------------------------------------------------------------------
__global__ __launch_bounds__(256) void gcn_scatter1(
    const int* __restrict__ src, const int* __restrict__ dst,
    const float* __restrict__ ew, const float* __restrict__ sup,
    float* __restrict__ agg) {
  int e = blockIdx.x * 8 + (threadIdx.x >> 5);
  if (e >= N_EDGES) return;
  int lane = threadIdx.x & 31;
  int s = __builtin_nontemporal_load(&src[e]);
  int d = __builtin_nontemporal_load(&dst[e]);
  float w = __builtin_nontemporal_load(&ew[e]);
  float4 v = ((const float4*)(sup + (size_t)s * NHID))[lane];
  float* ap = agg + (size_t)d * NHID + lane * 4;
  atomicAdd(ap + 0, v.x * w);
  atomicAdd(ap + 1, v.y * w);
  atomicAdd(ap + 2, v.z * w);
  atomicAdd(ap + 3, v.w * w);
}

// ---------------------------------------------------------------------------
// Scatter layer 2: one wave per edge; 40 features = 32 lanes + 8 extra.
// ---------------------------------------------------------------------------
__global__ __launch_bounds__(256) void gcn_scatter2(
    const int* __restrict__ src, const int* __restrict__ dst,
    const float* __restrict__ ew, const float* __restrict__ sup,
    float* __restrict__ agg) {
  int e = blockIdx.x * 8 + (threadIdx.x >> 5);
  if (e >= N_EDGES) return;
  int lane = threadIdx.x & 31;
  int s = __builtin_nontemporal_load(&src[e]);
  int d = __builtin_nontemporal_load(&dst[e]);
  float w = __builtin_nontemporal_load(&ew[e]);
  const float* sp = sup + (size_t)s * NCLASS;
  float* ap = agg + (size_t)d * NCLASS;
  atomicAdd(ap + lane, sp[lane] * w);
  if (lane < 8) atomicAdd(ap + 32 + lane, sp[32 + lane] * w);
}

// ---------------------------------------------------------------------------
// log_softmax over 40 classes; one wave32 per node, shfl_xor reductions.
// agg2 is last-use -> NT loads; output is never re-read -> NT stores.
// ---------------------------------------------------------------------------
__global__ __launch_bounds__(256) void gcn_logsoftmax(
    const float* __restrict__ agg2, float* __restrict__ out) {
  int node = blockIdx.x * 8 + (threadIdx.x >> 5);
  if (node >= N_NODES) return;
  int lane = threadIdx.x & 31;
  const float* p = agg2 + (size_t)node * NCLASS;
  float v0 = __builtin_nontemporal_load(&p[lane]);
  float v1 = (lane < 8) ? __builtin_nontemporal_load(&p[32 + lane]) : -INFINITY;

  float m = fmaxf(v0, v1);
#pragma unroll
  for (int off = 16; off > 0; off >>= 1) m = fmaxf(m, __shfl_xor(m, off, 32));

  float s = __expf(v0 - m) + ((lane < 8) ? __expf(v1 - m) : 0.0f);
#pragma unroll
  for (int off = 16; off > 0; off >>= 1) s += __shfl_xor(s, off, 32);

  float ls = __logf(s);
  __builtin_nontemporal_store(v0 - m - ls, &out[(size_t)node * NCLASS + lane]);
  if (lane < 8)
    __builtin_nontemporal_store(v1 - m - ls,
                                &out[(size_t)node * NCLASS + 32 + lane]);
}

// ---------------------------------------------------------------------------
extern "C" void kernel_launch(void* const* d_in, const int* in_sizes, int n_in,
                              void* d_out, int out_size, void* d_ws,
                              size_t ws_size, hipStream_t stream) {
  const float* x = (const float*)d_in[0];
  const float* W1 = (const float*)d_in[1];
  const float* b1 = (const float*)d_in[2];
  const float* W2 = (const float*)d_in[3];
  const float* b2 = (const float*)d_in[4];
  const int* ei = (const int*)d_in[5];
  const float* ew = (const float*)d_in[6];
  const int* src = ei;            // edge_index row 0
  const int* dst = ei + N_EDGES;  // edge_index row 1

  float* ws = (float*)d_ws;
  float* support1 = ws;                               // 100000*128
  float* agg1 = support1 + (size_t)N_NODES * NHID;    // 100000*128
  float* support2 = agg1 + (size_t)N_NODES * NHID;    // 100000*40
  float* agg2 = support2 + (size_t)N_NODES * NCLASS;  // 100000*40
  float* out = (float*)d_out;

  // Layer 1
  gcn_gemm1<<<N_NODES / 16, 256, 0, stream>>>(x, W1, support1);
  long tot1 = (long)N_NODES * NHID;
  gcn_bias_init<<<(int)((tot1 + 255) / 256), 256, 0, stream>>>(b1, agg1, NHID,
                                                               tot1);
  gcn_scatter1<<<N_EDGES / 8, 256, 0, stream>>>(src, dst, ew, support1, agg1);

  // Layer 2 (ReLU fused into GEMM2 A-stage)
  gcn_gemm2<<<N_NODES / 16, 96, 0, stream>>>(agg1, W2, support2);
  long tot2 = (long)N_NODES * NCLASS;
  gcn_bias_init<<<(int)((tot2 + 255) / 256), 256, 0, stream>>>(b2, agg2,
                                                               NCLASS, tot2);
  gcn_scatter2<<<N_EDGES / 8, 256, 0, stream>>>(src, dst, ew, support2, agg2);

  // log_softmax
  gcn_logsoftmax<<<(N_NODES + 7) / 8, 256, 0, stream>>>(agg2, out);
}